// PyramidROIAlign_55439437857593
// MI455X (gfx1250) — compile-verified
//
#include <hip/hip_runtime.h>
#include <hip/hip_bf16.h>

// PyramidROIAlign for MI455X (gfx1250, wave32).
// One wave per (roi, py, px) sample. The 4-corner bilinear blend is mapped
// onto V_WMMA_F32_16X16X4_F32: A(16x4) = the sample's 4 bilinear weights
// replicated into all 16 rows, B(4x16) = 4 corner values for a 16-channel
// tile, D(16x16) -> every row is the blended 16-channel output tile.
//
// ISA VGPR layouts used (CDNA5 ISA §7.12.2):
//   A 16x4 f32 (2 VGPRs): v0 = K0 (lanes 0-15) / K2 (lanes 16-31), v1 = K1/K3
//   B 4x16 f32 (2 VGPRs): row-striped across lanes, halves split K:
//                          lane n<16 holds {B[0][n],B[1][n]}, lane n+16 {B[2][n],B[3][n]}
//   D 16x16 f32 (8 VGPRs): every row identical here; all 32 lanes of d[0]
//                          hold out[c0 + (lane&15)], so a full-wave store just
//                          writes duplicate identical values (no EXEC churn).
//
// Memory schedule: all 32 corner loads for the 256-channel sample are issued
// up-front (in-order returns -> compiler emits partial s_wait_loadcnt), then
// 16 WMMAs + stores drain them. All tile offsets are immediates.

typedef __attribute__((ext_vector_type(2))) float v2f;
typedef __attribute__((ext_vector_type(8))) float v8f;

#define PH   7
#define PW   7
#define CCH  256
#define NTILE (CCH / 16)
#define WAVES_PER_BLOCK 8

__global__ __launch_bounds__(256)
void pyramid_roi_align_wmma(const float* __restrict__ boxes,
                            const float* __restrict__ image_meta,
                            const float* __restrict__ p2,
                            const float* __restrict__ p3,
                            const float* __restrict__ p4,
                            const float* __restrict__ p5,
                            float* __restrict__ out,
                            int B, int N)
{
    const int lane = threadIdx.x & 31;
    const int wave = threadIdx.x >> 5;
    const int s    = blockIdx.x * WAVES_PER_BLOCK + wave;   // sample = m*49 + py*7 + px
    const int total = B * N * PH * PW;
    if (s >= total) return;           // wave-uniform: EXEC stays all-1 below

    const int m   = s / (PH * PW);    // flat ROI index (b*N + n)
    const int pix = s - m * (PH * PW);
    const int py  = pix / PW;
    const int px  = pix - py * PW;
    const int b   = m / N;

    // ---- box + level selection (wave-uniform scalar math) ----
    const float y1 = boxes[m * 4 + 0];
    const float x1 = boxes[m * 4 + 1];
    const float y2 = boxes[m * 4 + 2];
    const float x2 = boxes[m * 4 + 3];
    const float h  = y2 - y1;
    const float w  = x2 - x1;

    const float area  = image_meta[4] * image_meta[5];   // row 0, cols 4,5
    const float scale = sqrtf(fmaxf(h * w, 1e-12f));
    const float roi_level = log2f(scale / (224.0f / sqrtf(area)));
    int lvl = 4 + (int)rintf(roi_level);                 // rintf == RNE == jnp.round
    lvl = min(5, max(2, lvl));

    const float* fmap;
    int H;
    switch (lvl) {
        case 2:  fmap = p2; H = 256; break;
        case 3:  fmap = p3; H = 128; break;
        case 4:  fmap = p4; H = 64;  break;
        default: fmap = p5; H = 32;  break;
    }

    // ---- bilinear sample position ----
    const float Hm1 = (float)(H - 1);
    const float ty  = (float)py * (1.0f / (float)(PH - 1));
    const float tx  = (float)px * (1.0f / (float)(PW - 1));
    const float ys  = (y1 + ty * h) * Hm1;
    const float xs  = (x1 + tx * w) * Hm1;
    const float fy  = floorf(ys);
    const float fx  = floorf(xs);
    const float ly  = ys - fy;
    const float lx  = xs - fx;
    const int y0  = min(H - 1, max(0, (int)fy));
    const int y1i = min(H - 1, max(0, (int)fy + 1));
    const int x0  = min(H - 1, max(0, (int)fx));
    const int x1i = min(H - 1, max(0, (int)fx + 1));

    // bilinear weights: k0=(y0,x0) k1=(y0,x1) k2=(y1,x0) k3=(y1,x1)
    const float w00 = (1.0f - ly) * (1.0f - lx);
    const float w01 = (1.0f - ly) * lx;
    const float w10 = ly * (1.0f - lx);
    const float w11 = ly * lx;

    // corner row base offsets (in floats); fmap layout [B,H,W,C]
    const size_t plane = (size_t)b * (size_t)H * (size_t)H;
    const size_t off00 = (plane + (size_t)y0  * H + x0 ) * CCH;
    const size_t off01 = (plane + (size_t)y0  * H + x1i) * CCH;
    const size_t off10 = (plane + (size_t)y1i * H + x0 ) * CCH;
    const size_t off11 = (plane + (size_t)y1i * H + x1i) * CCH;

    // ---- A matrix: weights replicated into all 16 rows ----
    const bool hi = (lane >= 16);
    v2f a;
    a.x = hi ? w10 : w00;   // K0 / K2
    a.y = hi ? w11 : w01;   // K1 / K3

    const int n = lane & 15;
    const float* __restrict__ srcA = fmap + (hi ? off10 : off00) + n;
    const float* __restrict__ srcB = fmap + (hi ? off11 : off01) + n;
    float* __restrict__ dst = out + (size_t)s * CCH + n;

    // ---- issue all 32 corner loads for the sample (offsets become immediates) ----
    v2f bmv[NTILE];
#pragma unroll
    for (int t = 0; t < NTILE; ++t) {
        bmv[t].x = srcA[t * 16];    // 64B-coalesced per half-wave
        bmv[t].y = srcB[t * 16];
    }

    // ---- 16 WMMAs + full-wave stores (lanes 16-31 write duplicate values) ----
#pragma unroll
    for (int t = 0; t < NTILE; ++t) {
        v8f c = {};
        // 8 args: (neg_a, A, neg_b, B, c_mod, C, reuse_a, reuse_b)
        v8f d = __builtin_amdgcn_wmma_f32_16x16x4_f32(
            false, a, false, bmv[t], (short)0, c, false, false);
        dst[t * 16] = d[0];         // all rows of D identical -> benign dup store
    }
}

extern "C" void kernel_launch(void* const* d_in, const int* in_sizes, int n_in,
                              void* d_out, int out_size, void* d_ws, size_t ws_size,
                              hipStream_t stream) {
    const float* boxes = (const float*)d_in[0];   // [B,N,4]
    const float* meta  = (const float*)d_in[1];   // [B,93]
    const float* p2    = (const float*)d_in[2];   // [B,256,256,256]
    const float* p3    = (const float*)d_in[3];   // [B,128,128,256]
    const float* p4    = (const float*)d_in[4];   // [B, 64, 64,256]
    const float* p5    = (const float*)d_in[5];   // [B, 32, 32,256]
    float* out = (float*)d_out;                   // [B,N,7,7,256]

    const int B = in_sizes[1] / 93;
    const int M = in_sizes[0] / 4;                // B*N
    const int N = M / B;

    const int total_samples = M * PH * PW;        // one wave32 each
    const int blocks = (total_samples + WAVES_PER_BLOCK - 1) / WAVES_PER_BLOCK;

    pyramid_roi_align_wmma<<<blocks, WAVES_PER_BLOCK * 32, 0, stream>>>(
        boxes, meta, p2, p3, p4, p5, out, B, N);
}